// MANO_81982335746323
// MI455X (gfx1250) — compile-verified
//
#include <hip/hip_runtime.h>
#include <hip/hip_bf16.h>

typedef __attribute__((ext_vector_type(2))) float v2f;
typedef __attribute__((ext_vector_type(8))) float v8f;

#define NVERTS 778
#define NJ     16
#define C3     2334   // 778*3
#define PFPAD  136    // 135 padded to multiple of 4
#define BPAD   12     // 10 padded to multiple of 4
#define NTILE  4      // column tiles per wave in the WMMA GEMM

// ---------------------------------------------------------------------------
// Kernel 1: batch Rodrigues. One thread per (n, joint).
// Writes Rs (output region 3) and zero-padded pose_feature rows (N x 136).
// ---------------------------------------------------------------------------
__global__ void k_rodrigues(const float* __restrict__ theta,
                            float* __restrict__ rs_out,   // N*16*9
                            float* __restrict__ pf,       // N*136
                            int N)
{
    int t = blockIdx.x * blockDim.x + threadIdx.x;
    if (t >= N * NJ) return;
    int n = t >> 4;
    int j = t & 15;
    const float* th = theta + (size_t)n * 48 + j * 3;
    float tx = th[0], ty = th[1], tz = th[2];
    float ax = tx + 1e-8f, ay = ty + 1e-8f, az = tz + 1e-8f;
    float angle = sqrtf(ax * ax + ay * ay + az * az);
    float inv   = 1.0f / angle;
    float half  = 0.5f * angle;
    float w = cosf(half);
    float s = sinf(half);
    float qx = s * tx * inv, qy = s * ty * inv, qz = s * tz * inv;
    float qn = 1.0f / sqrtf(w * w + qx * qx + qy * qy + qz * qz);
    w *= qn; qx *= qn; qy *= qn; qz *= qn;
    float w2 = w * w,  x2 = qx * qx, y2 = qy * qy, z2 = qz * qz;
    float wx = w * qx, wy = w * qy,  wz = w * qz;
    float xy = qx * qy, xz = qx * qz, yz = qy * qz;
    float R[9];
    R[0] = w2 + x2 - y2 - z2; R[1] = 2.f * xy - 2.f * wz; R[2] = 2.f * wy + 2.f * xz;
    R[3] = 2.f * wz + 2.f * xy; R[4] = w2 - x2 + y2 - z2; R[5] = 2.f * yz - 2.f * wx;
    R[6] = 2.f * xz - 2.f * wy; R[7] = 2.f * wx + 2.f * yz; R[8] = w2 - x2 - y2 + z2;

    float* ro = rs_out + (size_t)t * 9;
    #pragma unroll
    for (int i = 0; i < 9; i++) ro[i] = R[i];

    if (j > 0) {
        float* p = pf + (size_t)n * PFPAD + (j - 1) * 9;
        #pragma unroll
        for (int i = 0; i < 9; i++)
            p[i] = R[i] - ((i == 0 || i == 4 || i == 8) ? 1.f : 0.f);
    } else {
        pf[(size_t)n * PFPAD + 135] = 0.f;  // zero K-pad column
    }
}

// ---------------------------------------------------------------------------
// Pack kernels: build zero-padded operands so the GEMM inner loop has NO
// conditional loads.
//   k_pack_beta : beta (N x 10) -> betap (N x 12), cols 10..11 = 0
//   k_pack_pairs: B (K x 2334)  -> Bp (Kpad/2 x 2334 x 2), K-pad rows = 0,
//                 pair-interleaved so a lane's v2f B fragment is ONE b64 load.
// ---------------------------------------------------------------------------
__global__ void k_pack_beta(const float* __restrict__ beta,
                            float* __restrict__ betap, int N)
{
    int gid = blockIdx.x * blockDim.x + threadIdx.x;
    if (gid >= N * BPAD) return;
    int n = gid / BPAD;
    int k = gid - n * BPAD;
    betap[gid] = (k < 10) ? beta[(size_t)n * 10 + k] : 0.f;
}

__global__ void k_pack_pairs(const float* __restrict__ src, int K,
                             float* __restrict__ dst, int npairs)
{
    int gid = blockIdx.x * blockDim.x + threadIdx.x;
    if (gid >= npairs * C3) return;
    int p   = gid / C3;
    int col = gid - p * C3;
    int k0 = 2 * p, k1 = 2 * p + 1;
    dst[(size_t)gid * 2 + 0] = (k0 < K) ? src[(size_t)k0 * C3 + col] : 0.f;
    dst[(size_t)gid * 2 + 1] = (k1 < K) ? src[(size_t)k1 * C3 + col] : 0.f;
}

// ---------------------------------------------------------------------------
// fp32 WMMA GEMM  C[Nb x 2334] = A[Nb x Kpad] * B[Kpad x 2334] (+epilogue).
// One wave per 16 x (16*NTILE) C-strip. Software-pipelined: step ks+1's
// fragments (1 A b64 + NTILE B b64) are issued BEFORE step ks's four
// independent V_WMMA_F32_16X16X4_F32 ops consume their fragments, so the
// compiler can wait on loadcnt<=NTILE+1 instead of draining to 0 per step.
// Fragment layouts per ISA 7.12.2; padded operands -> no conditional loads,
// EXEC stays all-ones around the WMMAs.
// ---------------------------------------------------------------------------
__global__ void k_gemm_wmma(const float* __restrict__ Ap, int lda, int ksteps,
                            const float* __restrict__ Bp,        // paired layout
                            const float* __restrict__ addvec,    // 2334 or null
                            const float* __restrict__ addmat,    // Nb x 2334 or null
                            float* __restrict__ Cout)            // Nb x 2334
{
    int lane = threadIdx.x;          // 0..31 (one wave)
    int half = lane >> 4;            // 0: K={0,1}, 1: K={2,3} within a step
    int l16  = lane & 15;
    int row  = blockIdx.x * 16 + l16;

    const float* arow = Ap + (size_t)row * lda + half * 2;

    int cols[NTILE];
    const float* bptr[NTILE];
    #pragma unroll
    for (int t = 0; t < NTILE; t++) {
        int col  = (blockIdx.y * NTILE + t) * 16 + l16;
        int colc = col < (C3 - 1) ? col : (C3 - 1);
        cols[t]  = col;
        // pair p = 2*ks + half; element = (p*C3 + colc)*2
        bptr[t]  = Bp + ((size_t)half * C3 + colc) * 2;
    }

    v8f acc[NTILE];
    #pragma unroll
    for (int t = 0; t < NTILE; t++) acc[t] = (v8f){};

    // ---- prime: fragments for step 0 ----
    v2f a_cur = *(const v2f*)(arow);
    v2f b_cur[NTILE];
    #pragma unroll
    for (int t = 0; t < NTILE; t++) b_cur[t] = *(const v2f*)(bptr[t]);

    // ---- steady state: prefetch ks+1, consume ks ----
    for (int ks = 0; ks < ksteps - 1; ++ks) {
        v2f a_nxt = *(const v2f*)(arow + (ks + 1) * 4);
        v2f b_nxt[NTILE];
        #pragma unroll
        for (int t = 0; t < NTILE; t++)
            b_nxt[t] = *(const v2f*)(bptr[t] + (size_t)(ks + 1) * 4 * C3);

        #pragma unroll
        for (int t = 0; t < NTILE; t++)
            acc[t] = __builtin_amdgcn_wmma_f32_16x16x4_f32(
                         false, a_cur, false, b_cur[t], (short)0, acc[t], false, false);

        a_cur = a_nxt;
        #pragma unroll
        for (int t = 0; t < NTILE; t++) b_cur[t] = b_nxt[t];
    }

    // ---- drain: last step ----
    #pragma unroll
    for (int t = 0; t < NTILE; t++)
        acc[t] = __builtin_amdgcn_wmma_f32_16x16x4_f32(
                     false, a_cur, false, b_cur[t], (short)0, acc[t], false, false);

    // ---- epilogue ----
    #pragma unroll
    for (int t = 0; t < NTILE; t++) {
        int col = cols[t];
        if (col < C3) {
            #pragma unroll
            for (int i = 0; i < 8; i++) {
                int r = blockIdx.x * 16 + i + half * 8;  // C VGPR i -> rows i / i+8
                float v = acc[t][i];
                if (addvec) v += addvec[col];
                if (addmat) v += addmat[(size_t)r * C3 + col];
                Cout[(size_t)r * C3 + col] = v;
            }
        }
    }
}

// ---------------------------------------------------------------------------
// J = einsum('nvc,vj->njc'). One block per body; stage the body's 2334 floats
// in LDS, then 48 threads each do a 778-long reduction.
// ---------------------------------------------------------------------------
__global__ void k_regress(const float* __restrict__ src,   // N x 2334
                          const float* __restrict__ Jreg,  // 778 x 16
                          float* __restrict__ dst)         // N x 48
{
    __shared__ float sh[C3];
    int n = blockIdx.x;
    const float* s = src + (size_t)n * C3;
    for (int i = threadIdx.x; i < C3; i += blockDim.x) sh[i] = s[i];
    __syncthreads();
    int t = threadIdx.x;
    if (t < 48) {
        int j = t / 3, c = t % 3;
        float acc = 0.f;
        for (int v = 0; v < NVERTS; ++v)
            acc += sh[v * 3 + c] * Jreg[v * NJ + j];
        dst[(size_t)n * 48 + j * 3 + c] = acc;
    }
}

// ---------------------------------------------------------------------------
// Kinematic chain -> A matrices (N x 16 x 4x4). One thread per body.
// MANO chains: root 0, then (1..3)(4..6)(7..9)(10..12)(13..15): only one
// running transform live -> no spills.
// ---------------------------------------------------------------------------
__device__ __forceinline__ void emitA(float* __restrict__ Ao, int i,
                                      const float* R, const float* t,
                                      const float* Ji)
{
    float* a = Ao + i * 16;
    #pragma unroll
    for (int r = 0; r < 3; r++) {
        a[r * 4 + 0] = R[r * 3 + 0];
        a[r * 4 + 1] = R[r * 3 + 1];
        a[r * 4 + 2] = R[r * 3 + 2];
        a[r * 4 + 3] = t[r] - (R[r * 3 + 0] * Ji[0] + R[r * 3 + 1] * Ji[1] + R[r * 3 + 2] * Ji[2]);
    }
    a[12] = 0.f; a[13] = 0.f; a[14] = 0.f; a[15] = 1.f;
}

__global__ void k_chain(const float* __restrict__ rs,   // N*16*9
                        const float* __restrict__ Jw,   // N*48
                        float* __restrict__ Aw,         // N*256
                        int N)
{
    int n = blockIdx.x * blockDim.x + threadIdx.x;
    if (n >= N) return;
    const float* Rn = rs + (size_t)n * 144;
    const float* Jn = Jw + (size_t)n * 48;
    float* Ao = Aw + (size_t)n * 256;

    // root: Rs[0] @ ROT_X  (ROT_X = diag(1,-1,-1) -> negate cols 1,2)
    float R0[9], t0[3];
    #pragma unroll
    for (int r = 0; r < 3; r++) {
        R0[r * 3 + 0] =  Rn[r * 3 + 0];
        R0[r * 3 + 1] = -Rn[r * 3 + 1];
        R0[r * 3 + 2] = -Rn[r * 3 + 2];
    }
    t0[0] = Jn[0]; t0[1] = Jn[1]; t0[2] = Jn[2];
    emitA(Ao, 0, R0, t0, Jn);

    for (int c = 0; c < 5; c++) {
        float Rc[9], tc[3];
        #pragma unroll
        for (int q = 0; q < 9; q++) Rc[q] = R0[q];
        tc[0] = t0[0]; tc[1] = t0[1]; tc[2] = t0[2];
        int prev = 0;
        #pragma unroll
        for (int s = 0; s < 3; s++) {
            int i = c * 3 + s + 1;
            const float* Ri = Rn + i * 9;
            float ti[3] = { Jn[i * 3 + 0] - Jn[prev * 3 + 0],
                            Jn[i * 3 + 1] - Jn[prev * 3 + 1],
                            Jn[i * 3 + 2] - Jn[prev * 3 + 2] };
            float Rm[9], tm[3];
            #pragma unroll
            for (int r = 0; r < 3; r++) {
                #pragma unroll
                for (int q = 0; q < 3; q++)
                    Rm[r * 3 + q] = Rc[r * 3 + 0] * Ri[0 * 3 + q]
                                  + Rc[r * 3 + 1] * Ri[1 * 3 + q]
                                  + Rc[r * 3 + 2] * Ri[2 * 3 + q];
                tm[r] = Rc[r * 3 + 0] * ti[0] + Rc[r * 3 + 1] * ti[1] + Rc[r * 3 + 2] * ti[2] + tc[r];
            }
            #pragma unroll
            for (int q = 0; q < 9; q++) Rc[q] = Rm[q];
            tc[0] = tm[0]; tc[1] = tm[1]; tc[2] = tm[2];
            emitA(Ao, i, Rc, tc, Jn + i * 3);
            prev = i;
        }
    }
}

// ---------------------------------------------------------------------------
// Blend skinning. Grid (vert-tile, n); block stages A[n] (256 floats) in LDS
// once, then each thread handles one vertex.
// ---------------------------------------------------------------------------
__global__ void k_lbs(const float* __restrict__ V,       // N x 2334 (v_posed)
                      const float* __restrict__ Aw,      // N x 256
                      const float* __restrict__ weights, // 778 x 16
                      float* __restrict__ verts)         // N x 778 x 3
{
    __shared__ float Ash[256];
    int n   = blockIdx.y;
    int tid = threadIdx.x;
    Ash[tid] = Aw[(size_t)n * 256 + tid];     // blockDim.x == 256
    __syncthreads();

    int v = blockIdx.x * blockDim.x + tid;
    if (v >= NVERTS) return;
    const float* vp = V + (size_t)n * C3 + v * 3;
    float x = vp[0], y = vp[1], z = vp[2];
    const float* wv = weights + v * NJ;
    float ox = 0.f, oy = 0.f, oz = 0.f;
    #pragma unroll
    for (int j = 0; j < NJ; j++) {
        float w = wv[j];
        const float* a = Ash + j * 16;
        ox += w * (a[0] * x + a[1] * y + a[2]  * z + a[3]);
        oy += w * (a[4] * x + a[5] * y + a[6]  * z + a[7]);
        oz += w * (a[8] * x + a[9] * y + a[10] * z + a[11]);
    }
    float* o = verts + ((size_t)n * NVERTS + v) * 3;
    o[0] = ox; o[1] = oy; o[2] = oz;
}

// ---------------------------------------------------------------------------
extern "C" void kernel_launch(void* const* d_in, const int* in_sizes, int n_in,
                              void* d_out, int out_size, void* d_ws, size_t ws_size,
                              hipStream_t stream)
{
    const float* beta       = (const float*)d_in[0];   // N x 10
    const float* theta      = (const float*)d_in[1];   // N x 48
    const float* v_template = (const float*)d_in[2];   // 778 x 3 (flat 2334)
    const float* shapedirs  = (const float*)d_in[3];   // 10 x 2334
    const float* Jreg       = (const float*)d_in[4];   // 778 x 16
    const float* posedirs   = (const float*)d_in[5];   // 135 x 2334
    const float* weights    = (const float*)d_in[6];   // 778 x 16

    const int N = in_sizes[0] / 10;                    // 4096

    // Output regions: verts | joints | Rs (flat, in return order)
    float* out    = (float*)d_out;
    float* verts  = out;                               // N*778*3
    float* joints = verts + (size_t)N * NVERTS * 3;    // N*48
    float* rs     = joints + (size_t)N * 48;           // N*144

    // Workspace layout
    float* ws    = (float*)d_ws;
    float* pf    = ws;                                 // N*136 (padded pose_feature)
    float* V     = pf    + (size_t)N * PFPAD;          // N*2334 (v_shaped -> v_posed)
    float* Jw    = V     + (size_t)N * C3;             // N*48
    float* Aw    = Jw    + (size_t)N * 48;             // N*256
    float* betap = Aw    + (size_t)N * 256;            // N*12
    float* Bs    = betap + (size_t)N * BPAD;           // (12/2)*2334*2  shapedirs pairs
    float* Bpz   = Bs    + (size_t)(BPAD / 2) * C3 * 2; // (136/2)*2334*2 posedirs pairs

    dim3 gtile(N / 16, (C3 / 16 + NTILE) / NTILE);     // 256 x 37 wave strips

    // 1. Rodrigues -> Rs output + padded pose_feature
    k_rodrigues<<<(N * NJ + 255) / 256, 256, 0, stream>>>(theta, rs, pf, N);

    // 1b. pack padded operands (removes all conditional loads from the GEMMs)
    k_pack_beta<<<(N * BPAD + 255) / 256, 256, 0, stream>>>(beta, betap, N);
    k_pack_pairs<<<((BPAD / 2) * C3 + 255) / 256, 256, 0, stream>>>(shapedirs, 10, Bs, BPAD / 2);
    k_pack_pairs<<<((PFPAD / 2) * C3 + 255) / 256, 256, 0, stream>>>(posedirs, 135, Bpz, PFPAD / 2);

    // 2. v_shaped = beta @ shapedirs + v_template   (WMMA f32, Kpad=12)
    k_gemm_wmma<<<gtile, 32, 0, stream>>>(betap, BPAD, BPAD / 4, Bs,
                                          v_template, nullptr, V);

    // 3. J = regress(v_shaped)
    k_regress<<<N, 64, 0, stream>>>(V, Jreg, Jw);

    // 4. v_posed = pose_feature @ posedirs + v_shaped   (WMMA f32, Kpad=136, in place)
    k_gemm_wmma<<<gtile, 32, 0, stream>>>(pf, PFPAD, PFPAD / 4, Bpz,
                                          nullptr, V, V);

    // 5. kinematic chain -> A matrices
    k_chain<<<(N + 63) / 64, 64, 0, stream>>>(rs, Jw, Aw, N);

    // 6. linear blend skinning -> verts (A[n] staged in LDS per block)
    dim3 glbs((NVERTS + 255) / 256, N);
    k_lbs<<<glbs, 256, 0, stream>>>(V, Aw, weights, verts);

    // 7. joints = regress(verts)
    k_regress<<<N, 64, 0, stream>>>(verts, Jreg, joints);
}